// Hawk_27015344292123
// MI455X (gfx1250) — compile-verified
//
#include <hip/hip_runtime.h>
#include <hip/hip_bf16.h>
#include <math.h>
#include <stdint.h>

// ---------------------------------------------------------------------------
// Hawk (RG-LRU) forward for MI455X / gfx1250.
//   B=4, T=4096, DIM=E=1024, conv K=4.
// GEMMs: V_WMMA_F32_16X16X4_F32 (f32 matrix pipe, matches reference
// precision) with double-buffered GLOBAL_LOAD_ASYNC_TO_LDS_B128 staging.
// Recurrence: 3-phase chunked parallel scan.
// ---------------------------------------------------------------------------

#define HB 4
#define HT 4096
#define HE 1024
#define HM (HB * HT)          // 16384 rows for every GEMM
#define NCH 32                // chunks per sequence
#define CL  (HT / NCH)        // 128 steps per chunk

typedef __attribute__((ext_vector_type(2))) float v2f;
typedef __attribute__((ext_vector_type(8))) float v8f;

// ---------------------------------------------------------------------------
// WMMA GEMM:  C[M,N] = A[M,K] * W[N,K]^T (+ bias[N])
// Block: 256 threads (8 waves). Block tile 128(M) x 64(N). K tile = 32.
// Wave tile 32x32 = 2x2 tiles of 16x16 -> 4 x v8f accumulators.
// LDS row stride padded to 36 floats: 16 fragment rows map to 16 distinct
// banks (64-bank LDS), float2/float4 alignment preserved.
// Global->LDS staging uses async copies (ASYNCcnt) with double buffering:
// the copy of K-tile k+1 overlaps the WMMA work on K-tile k.
// ---------------------------------------------------------------------------
#define LDT 36
#define SH_A (128 * LDT)
#define SH_B (64 * LDT)

__device__ __forceinline__ void hawk_async_b128(float* lds_dst, const float* gsrc) {
    // LDS address = low 32 bits of the generic pointer (HW truncates addr[31:0]).
    uint32_t lds = (uint32_t)(uintptr_t)lds_dst;
    uint64_t gp  = (uint64_t)(uintptr_t)gsrc;
    asm volatile("global_load_async_to_lds_b128 %0, %1, off"
                 :: "v"(lds), "v"(gp) : "memory");
}

__device__ __forceinline__ void hawk_wait_async0() {
    asm volatile("s_wait_asynccnt 0" ::: "memory");
}

// Issue one K-tile's worth of async copies (6 x b128 per thread).
__device__ __forceinline__ void hawk_issue_tile(const float* __restrict__ A,
                                                const float* __restrict__ W,
                                                float* bufA, float* bufB,
                                                int mBase, int nBase,
                                                int K, int kb, int tid) {
    #pragma unroll
    for (int t = 0; t < 4; ++t) {            // A tile: 128 x 32 = 1024 float4
        int idx  = t * 256 + tid;
        int row  = idx >> 3;
        int col4 = (idx & 7) * 4;
        hawk_async_b128(&bufA[row * LDT + col4],
                        &A[(size_t)(mBase + row) * K + kb + col4]);
    }
    #pragma unroll
    for (int t = 0; t < 2; ++t) {            // W tile: 64 x 32 = 512 float4
        int idx  = t * 256 + tid;
        int row  = idx >> 3;
        int col4 = (idx & 7) * 4;
        hawk_async_b128(&bufB[row * LDT + col4],
                        &W[(size_t)(nBase + row) * K + kb + col4]);
    }
}

__global__ __launch_bounds__(256)
void hawk_wmma_gemm(const float* __restrict__ A,
                    const float* __restrict__ W,
                    const float* __restrict__ bias,
                    float* __restrict__ C,
                    int M, int N, int K) {
    __shared__ float shA[2][SH_A];
    __shared__ float shB[2][SH_B];

    const int tid  = threadIdx.x;
    const int lane = tid & 31;
    const int wave = tid >> 5;
    const int l15  = lane & 15;
    const int lh   = lane >> 4;          // 0: K pair {0,1}, 1: K pair {2,3}

    const int mBase = blockIdx.y * 128;
    const int nBase = blockIdx.x * 64;
    const int wm = (wave & 3) * 32;      // wave M offset in block tile
    const int wn = (wave >> 2) * 32;     // wave N offset in block tile

    v8f acc[2][2] = {};

    // prologue: stage K-tile 0
    hawk_issue_tile(A, W, shA[0], shB[0], mBase, nBase, K, 0, tid);
    hawk_wait_async0();
    __syncthreads();

    int cur = 0;
    for (int kb = 0; kb < K; kb += 32) {
        // kick off next tile into the other buffer (overlaps WMMA below)
        if (kb + 32 < K)
            hawk_issue_tile(A, W, shA[cur ^ 1], shB[cur ^ 1],
                            mBase, nBase, K, kb + 32, tid);

        const float* sA = shA[cur];
        const float* sB = shB[cur];

        // ---- 8 WMMA k-steps of 4 over the 32-wide K tile ----
        #pragma unroll
        for (int kk = 0; kk < 32; kk += 4) {
            const int kc = kk + 2 * lh;  // this lane-half's K pair
            // A fragments (16x4): lanes 0-15 = rows, VGPR pair = K, K+1
            v2f af0 = *(const v2f*)&sA[(wm +      l15) * LDT + kc];
            v2f af1 = *(const v2f*)&sA[(wm + 16 + l15) * LDT + kc];
            // B fragments (4x16 = Wtile^T): identical per-lane indexing
            v2f bf0 = *(const v2f*)&sB[(wn +      l15) * LDT + kc];
            v2f bf1 = *(const v2f*)&sB[(wn + 16 + l15) * LDT + kc];

            acc[0][0] = __builtin_amdgcn_wmma_f32_16x16x4_f32(
                false, af0, false, bf0, (short)0, acc[0][0], false, false);
            acc[0][1] = __builtin_amdgcn_wmma_f32_16x16x4_f32(
                false, af0, false, bf1, (short)0, acc[0][1], false, false);
            acc[1][0] = __builtin_amdgcn_wmma_f32_16x16x4_f32(
                false, af1, false, bf0, (short)0, acc[1][0], false, false);
            acc[1][1] = __builtin_amdgcn_wmma_f32_16x16x4_f32(
                false, af1, false, bf1, (short)0, acc[1][1], false, false);
        }

        hawk_wait_async0();   // next tile fully landed in LDS
        __syncthreads();      // (compiler also waits DScnt before the barrier)
        cur ^= 1;
    }

    // ---- epilogue: C/D layout -> VGPR r holds M = r + 8*lh, N = l15 ----
    #pragma unroll
    for (int j = 0; j < 2; ++j) {
        const int col = nBase + wn + j * 16 + l15;
        const float bv = bias ? bias[col] : 0.0f;
        #pragma unroll
        for (int i = 0; i < 2; ++i) {
            #pragma unroll
            for (int r = 0; r < 8; ++r) {
                const int row = mBase + wm + i * 16 + r + 8 * lh;
                C[(size_t)row * N + col] = acc[i][j][r] + bv;
            }
        }
    }
}

// ---------------------------------------------------------------------------
// -8 * softplus(forget_base[e]) precompute
// ---------------------------------------------------------------------------
__global__ void hawk_fb_coef(const float* __restrict__ fb, float* __restrict__ fbc) {
    int e = blockIdx.x * blockDim.x + threadIdx.x;
    if (e < HE) fbc[e] = -8.0f * log1pf(__expf(fb[e]));
}

// ---------------------------------------------------------------------------
// Causal depthwise conv1d, K=4, left pad 3:
//   xc[b,t,e] = b_conv[e] + sum_k w_conv[e,k] * xa[b, t-3+k, e]
// xa lives in proj[:, E:2E].
// ---------------------------------------------------------------------------
__global__ __launch_bounds__(256)
void hawk_dwconv(const float* __restrict__ proj,
                 const float* __restrict__ wconv,
                 const float* __restrict__ bconv,
                 float* __restrict__ xc) {
    int tid = blockIdx.x * blockDim.x + threadIdx.x;   // = (b*T+t)*E + e
    int e = tid & (HE - 1);
    int t = (tid / HE) & (HT - 1);
    int bt = tid / HE;                                  // b*T + t
    float acc = bconv[e];
    #pragma unroll
    for (int k = 0; k < 4; ++k) {
        int ts = t - 3 + k;
        if (ts >= 0)
            acc += wconv[e * 4 + k] *
                   proj[(size_t)(bt - 3 + k) * (2 * HE) + HE + e];
    }
    xc[tid] = acc;
}

// ---------------------------------------------------------------------------
// Recurrence math helpers
// ---------------------------------------------------------------------------
__device__ __forceinline__ float hawk_sigmoid(float x) {
    return 1.0f / (1.0f + __expf(-x));
}
__device__ __forceinline__ void hawk_alpha_u(float f, float inp, float xv,
                                             float coef, float& alpha, float& u) {
    alpha = __expf(coef * hawk_sigmoid(f));
    float beta = sqrtf(1.0f - alpha * alpha + 1e-6f);
    u = beta * hawk_sigmoid(inp) * xv;
}

// Phase A: per-chunk (prod(alpha), scan tail with h0=0).  tid = (b*NCH+c)*E + e
__global__ __launch_bounds__(256)
void hawk_chunk_scan(const float* __restrict__ g,
                     const float* __restrict__ xc,
                     const float* __restrict__ fbc,
                     float* __restrict__ chA,
                     float* __restrict__ chS) {
    int tid = blockIdx.x * blockDim.x + threadIdx.x;
    int e = tid & (HE - 1);
    int c = (tid / HE) & (NCH - 1);
    int b = tid / (HE * NCH);
    const float coef = fbc[e];
    float Ap = 1.0f, S = 0.0f;
    const int t0 = b * HT + c * CL;
    for (int i = 0; i < CL; ++i) {
        size_t gb = (size_t)(t0 + i) * (2 * HE);
        float alpha, u;
        hawk_alpha_u(g[gb + e], g[gb + HE + e],
                     xc[(size_t)(t0 + i) * HE + e], coef, alpha, u);
        Ap *= alpha;
        S = alpha * S + u;
    }
    chA[tid] = Ap;
    chS[tid] = S;
}

// Phase B: serial prefix over the 32 chunk summaries per (b,e)
__global__ __launch_bounds__(256)
void hawk_chunk_carry(const float* __restrict__ chA,
                      const float* __restrict__ chS,
                      float* __restrict__ carry) {
    int tid = blockIdx.x * blockDim.x + threadIdx.x;   // < B*E
    int e = tid & (HE - 1);
    int b = tid / HE;
    float h = 0.0f;
    for (int c = 0; c < NCH; ++c) {
        int idx = (b * NCH + c) * HE + e;
        carry[idx] = h;
        h = chA[idx] * h + chS[idx];
    }
}

// Phase C: replay chunk with carry-in, fuse y = gelu(gate) * h
__global__ __launch_bounds__(256)
void hawk_chunk_apply(const float* __restrict__ g,
                      const float* __restrict__ xc,
                      const float* __restrict__ proj,
                      const float* __restrict__ fbc,
                      const float* __restrict__ carry,
                      float* __restrict__ y) {
    int tid = blockIdx.x * blockDim.x + threadIdx.x;
    int e = tid & (HE - 1);
    int c = (tid / HE) & (NCH - 1);
    int b = tid / (HE * NCH);
    const float coef = fbc[e];
    float h = carry[tid];
    const int t0 = b * HT + c * CL;
    for (int i = 0; i < CL; ++i) {
        size_t gb = (size_t)(t0 + i) * (2 * HE);
        float alpha, u;
        hawk_alpha_u(g[gb + e], g[gb + HE + e],
                     xc[(size_t)(t0 + i) * HE + e], coef, alpha, u);
        h = alpha * h + u;
        float gate = proj[gb + e];  // first half of proj
        float gelu = 0.5f * gate * (1.0f + erff(gate * 0.70710678118654752f));
        y[(size_t)(t0 + i) * HE + e] = gelu * h;
    }
}

// ---------------------------------------------------------------------------
// Launch
// ---------------------------------------------------------------------------
extern "C" void kernel_launch(void* const* d_in, const int* in_sizes, int n_in,
                              void* d_out, int out_size, void* d_ws, size_t ws_size,
                              hipStream_t stream) {
    const float* x       = (const float*)d_in[0];   // [B,T,DIM]
    const float* w_in    = (const float*)d_in[1];   // [2E,DIM]
    const float* w_conv  = (const float*)d_in[2];   // [E,1,4]
    const float* b_conv  = (const float*)d_in[3];   // [E]
    const float* w_gates = (const float*)d_in[4];   // [2E,E]
    const float* b_gates = (const float*)d_in[5];   // [2E]
    const float* fbase   = (const float*)d_in[6];   // [E]
    const float* w_out   = (const float*)d_in[7];   // [DIM,E]
    float* out = (float*)d_out;                     // [B,T,DIM]

    // workspace layout (floats)
    float* ws    = (float*)d_ws;
    float* proj  = ws;                                  // [M, 2E]  128 MB
    float* xc    = ws + (size_t)HM * 2 * HE;            // [M, E]    64 MB
    float* g     = xc + (size_t)HM * HE;                // [M, 2E]  128 MB
    float* y     = g  + (size_t)HM * 2 * HE;            // [M, E]    64 MB
    float* fbc   = y  + (size_t)HM * HE;                // [E]
    float* chA   = fbc + HE;                            // [B,NCH,E]
    float* chS   = chA + (size_t)HB * NCH * HE;
    float* carry = chS + (size_t)HB * NCH * HE;

    dim3 blk(256);

    // 0) forget-base coefficient
    hawk_fb_coef<<<dim3((HE + 255) / 256), blk, 0, stream>>>(fbase, fbc);

    // 1) proj = x @ w_in^T             (M=16384, N=2048, K=1024)
    hawk_wmma_gemm<<<dim3(2 * HE / 64, HM / 128), blk, 0, stream>>>(
        x, w_in, nullptr, proj, HM, 2 * HE, HE);

    // 2) causal depthwise conv + bias
    hawk_dwconv<<<dim3((size_t)HM * HE / 256), blk, 0, stream>>>(
        proj, w_conv, b_conv, xc);

    // 3) g = xc @ w_gates^T + b_gates  (M=16384, N=2048, K=1024)
    hawk_wmma_gemm<<<dim3(2 * HE / 64, HM / 128), blk, 0, stream>>>(
        xc, w_gates, b_gates, g, HM, 2 * HE, HE);

    // 4) chunked parallel scan + gated output
    hawk_chunk_scan<<<dim3(HB * NCH * HE / 256), blk, 0, stream>>>(
        g, xc, fbc, chA, chS);
    hawk_chunk_carry<<<dim3(HB * HE / 256), blk, 0, stream>>>(chA, chS, carry);
    hawk_chunk_apply<<<dim3(HB * NCH * HE / 256), blk, 0, stream>>>(
        g, xc, proj, fbc, carry, y);

    // 5) out = y @ w_out^T             (M=16384, N=1024, K=1024)
    hawk_wmma_gemm<<<dim3(HE / 64, HM / 128), blk, 0, stream>>>(
        y, w_out, nullptr, out, HM, HE, HE);
}